// PPLAD_82480551952406
// MI455X (gfx1250) — compile-verified
//
#include <hip/hip_runtime.h>
#include <hip/hip_bf16.h>

// ---------------------------------------------------------------------------
// Problem constants (from the reference): B=128, L=8192, IN_FEAT=4, D_MODEL=256
// LOCAL_SIZE = {3,5,7}, GLOBAL_SIZE = {1,1,1}
// Output layout (flat, in return order), BL = B*L:
//   [ 0, 3BL)   local0  (B,L,3)
//   [ 3BL, 8BL) local1  (B,L,5)
//   [ 8BL,15BL) local2  (B,L,7)
//   [15BL,16BL) global0 (B,L,1)
//   [16BL,17BL) global1
//   [17BL,18BL) global2
//   [18BL,21BL) area_local 0..2  (B,L) each
//   [21BL,24BL) area_global 0..2 (B,L) each
//   [24BL,25BL) sigma
//   [25BL,26BL) sigma (again)
// ---------------------------------------------------------------------------

typedef float v2f __attribute__((ext_vector_type(2)));
typedef float v8f __attribute__((ext_vector_type(8)));

#define BL_TOTAL (128LL * 8192LL)
#define LW 8192

// ---------------------------------------------------------------------------
// Kernel 1: tiny-MLP via V_WMMA_F32_16X16X4_F32 (K=4 == IN_FEAT).
// Each wave handles 16 positions. Block = 256 threads = 8 waves = 128 positions.
//   A (16x4 f32):  lane(0..15)=rows, v[0]=K{0|2}, v[1]=K{1|3} per half
//   B (4x16 f32):  W1 slice, symmetric layout
//   C (16x16 f32): b1[n] broadcast down the M dimension
// After each of the 16 N-chunks: relu, scale by W2[n], accumulate per-lane.
// xor-shuffle (1,2,4,8) sums the 16 N-lanes inside each half -> s per row.
// ---------------------------------------------------------------------------
__global__ void __launch_bounds__(256) PPLAD_mlp_sigma_kernel(
    const float* __restrict__ in_x,   // (B*L, 4)
    const float* __restrict__ W1,     // (4, 256) row-major
    const float* __restrict__ b1,     // (256,)
    const float* __restrict__ W2,     // (256,)
    const float* __restrict__ b2,     // (1,)
    float* __restrict__ sig0,         // out + 24*BL
    float* __restrict__ sig1)         // out + 25*BL
{
    const int lane = threadIdx.x & 31;
    const int wave = threadIdx.x >> 5;
    const int half = lane >> 4;          // 0: lanes 0-15, 1: lanes 16-31
    const int nlo  = lane & 15;
    const long long P = (long long)blockIdx.x * 128 + (long long)wave * 16;

    // A-matrix: row M = nlo, this lane supplies K = 2*half, 2*half+1
    const float* arow = in_x + (P + nlo) * 4 + 2 * half;
    v2f A;
    A.x = arow[0];
    A.y = arow[1];

    float partial[8];
#pragma unroll
    for (int r = 0; r < 8; ++r) partial[r] = 0.0f;

#pragma unroll
    for (int c = 0; c < 16; ++c) {
        const int n = c * 16 + nlo;              // output-feature column
        v2f Bm;
        Bm.x = W1[(2 * half) * 256 + n];         // K = 2*half
        Bm.y = W1[(2 * half + 1) * 256 + n];     // K = 2*half + 1
        const float bias = b1[n];
        v8f C;
#pragma unroll
        for (int r = 0; r < 8; ++r) C[r] = bias;

        // D = A x B + C   (16x16 f32 accumulator, 8 VGPRs)
        v8f D = __builtin_amdgcn_wmma_f32_16x16x4_f32(
            /*neg_a=*/false, A, /*neg_b=*/false, Bm,
            /*c_mod=*/(short)0, C, /*reuse_a=*/false, /*reuse_b=*/false);

        const float w2v = W2[n];
#pragma unroll
        for (int r = 0; r < 8; ++r)
            partial[r] += fmaxf(D[r], 0.0f) * w2v;
    }

    // Reduce across the 16 N-lanes inside each half (halves hold M and M+8).
#pragma unroll
    for (int r = 0; r < 8; ++r) {
        float v = partial[r];
        v += __shfl_xor(v, 1, 32);
        v += __shfl_xor(v, 2, 32);
        v += __shfl_xor(v, 4, 32);
        v += __shfl_xor(v, 8, 32);
        partial[r] = v;
    }

    const float b2v = b2[0];
#pragma unroll
    for (int r = 0; r < 8; ++r) {
        const float s  = partial[r] + b2v;
        float sg = 1.0f / (1.0f + __expf(-5.0f * s)) + 1e-5f;   // sigmoid(5s)+1e-5
        sg = __expf(sg * 1.0986122886681098f) - 1.0f;           // 3^sg - 1
        if (nlo == r) {                                          // one writer / row
            const long long o = P + r + 8 * half;                // M = r + 8*half
            sig0[o] = sg;
            sig1[o] = sg;
        }
    }
}

// ---------------------------------------------------------------------------
// Kernel 2: per-position expansion. Reads sigma (4 MB), streams 96 MB of
// Gaussian kernel values + erf areas. Purely HBM-bound.
// ---------------------------------------------------------------------------
__global__ void __launch_bounds__(256) PPLAD_expand_kernel(
    const float* __restrict__ sigma,   // out + 24*BL
    float* out)
{
    const long long p = (long long)blockIdx.x * 256 + threadIdx.x;
    const int   l   = (int)(p % LW);
    const float pos = (float)l;
    const float Lmax = (float)(LW - 1);

    const float sig       = sigma[p];
    const float inv_norm  = 0.3989422804014327f / sig;        // 1/(sqrt(2pi)*sig)
    const float inv2s2    = 0.5f / (sig * sig);
    const float inv_ss2   = 1.0f / (sig * 1.41421356237309515f);

    const long long BL = BL_TOTAL;
    float* const outL0 = out;
    float* const outL1 = out + 3 * BL;
    float* const outL2 = out + 8 * BL;
    float* const outG  = out + 15 * BL;
    float* const outAL = out + 18 * BL;
    float* const outAG = out + 21 * BL;
    float* const outLoc[3] = { outL0, outL1, outL2 };

    const int LOCAL[3] = { 3, 5, 7 };

#pragma unroll
    for (int idx = 0; idx < 3; ++idx) {
        const int loc    = LOCAL[idx];
        const int num    = loc + 1;       // loc + GLOBAL_SIZE (=1)
        const int front1 = num / 2;
        const int front  = loc / 2;

        // k = 0 -> global_out (slice before the local window)
        {
            const float col = fminf(fmaxf(pos - (float)front1, 0.0f), Lmax);
            const float d   = fabsf(pos - col);
            outG[(long long)idx * BL + p] = inv_norm * __expf(-d * d * inv2s2);
        }
        // k = 1 .. num-1 -> local_out
#pragma unroll
        for (int k = 1; k < num; ++k) {
            const float col = fminf(fmaxf(pos - (float)front1 + (float)k, 0.0f), Lmax);
            const float d   = fabsf(pos - col);
            outLoc[idx][p * (long long)loc + (k - 1)] =
                inv_norm * __expf(-d * d * inv2s2);
        }
        // area_local = cdf(first) - cdf(last) = 0.5*(erf(a) - erf(b))
        {
            const float first = fminf(fmaxf(pos - (float)front, 0.0f), Lmax);
            const float last  = fminf(fmaxf(pos - (float)front + (float)(loc - 1), 0.0f), Lmax);
            const float al = 0.5f * (erff((first - pos) * inv_ss2) -
                                     erff((last  - pos) * inv_ss2));
            outAL[(long long)idx * BL + p] = al;
        }
        // area_global = -cdf(glast)
        {
            const float glast = fminf(fmaxf(pos - (float)front - 1.0f, 0.0f), Lmax);
            outAG[(long long)idx * BL + p] =
                -0.5f * (1.0f + erff((glast - pos) * inv_ss2));
        }
    }
}

// ---------------------------------------------------------------------------
extern "C" void kernel_launch(void* const* d_in, const int* in_sizes, int n_in,
                              void* d_out, int out_size, void* d_ws, size_t ws_size,
                              hipStream_t stream) {
    // setup_inputs order: x_in, in_x, W1, b1, W2, b2, in_size, in_num, op, it
    const float* in_x = (const float*)d_in[1];
    const float* W1   = (const float*)d_in[2];
    const float* b1   = (const float*)d_in[3];
    const float* W2   = (const float*)d_in[4];
    const float* b2   = (const float*)d_in[5];
    float* out = (float*)d_out;

    const long long BL = BL_TOTAL;          // 1,048,576 positions
    float* sig0 = out + 24 * BL;
    float* sig1 = out + 25 * BL;

    // Kernel 1: 8 waves/block, 16 positions/wave -> 128 positions/block
    const int blocksA = (int)(BL / 128);
    PPLAD_mlp_sigma_kernel<<<blocksA, 256, 0, stream>>>(in_x, W1, b1, W2, b2,
                                                        sig0, sig1);

    // Kernel 2: 1 position/thread
    const int blocksB = (int)(BL / 256);
    PPLAD_expand_kernel<<<blocksB, 256, 0, stream>>>(sig0, out);
}